// GNNPolicy_12412455486090
// MI455X (gfx1250) — compile-verified
//
#include <hip/hip_runtime.h>

typedef __attribute__((ext_vector_type(2))) float v2f;
typedef __attribute__((ext_vector_type(8))) float v8f;

#define N_NODES 51200
#define N_EDGES 819200
#define HIDDEN  256
#define N_GRAPHS 16

// ---------------------------------------------------------------------------
// Degree accumulation at edge targets (self-loop added later in rsqrt pass)
// ---------------------------------------------------------------------------
__global__ __launch_bounds__(256) void deg_kernel(const int* __restrict__ dst,
                                                  float* __restrict__ deg, int E) {
    int e = blockIdx.x * 256 + threadIdx.x;
    if (e < E) atomicAdd(&deg[dst[e]], 1.0f);
}

__global__ __launch_bounds__(256) void dis_kernel(float* __restrict__ deg, int N) {
    int n = blockIdx.x * 256 + threadIdx.x;
    if (n < N) deg[n] = rsqrtf(deg[n] + 1.0f);   // deg_inv_sqrt incl. self loop
}

// ---------------------------------------------------------------------------
// Input projection: x = coords @ W_in + b_in   ([N,2] @ [2,H])
// ---------------------------------------------------------------------------
__global__ __launch_bounds__(256) void input_proj_kernel(const float* __restrict__ coords,
                                                         const float* __restrict__ W_in,
                                                         const float* __restrict__ b_in,
                                                         float* __restrict__ x, int N) {
    int t = blockIdx.x * 256 + threadIdx.x;
    if (t >= N * HIDDEN) return;
    int n = t >> 8;
    int h = t & 255;
    x[t] = coords[2 * n] * W_in[h] + coords[2 * n + 1] * W_in[HIDDEN + h] + b_in[h];
}

// ---------------------------------------------------------------------------
// WMMA fp32 GEMM: Y[N,256] = X[N,256] @ W[256,256]  (+bias, +relu optional)
// Block: 512 threads = 16 waves. Block computes 16 rows x 256 cols.
// Each wave owns one 16x16 C tile, K loop in steps of 4 -> 64 WMMAs.
// Fragment layouts per CDNA5 ISA 7.12.2 (16x16x4 f32).
// ---------------------------------------------------------------------------
template <bool BIAS_RELU>
__global__ __launch_bounds__(512) void gemm_wmma_kernel(const float* __restrict__ X,
                                                        const float* __restrict__ W,
                                                        const float* __restrict__ bias,
                                                        float* __restrict__ Y) {
    __shared__ float Xs[16 * HIDDEN];            // 16 KB row slab

    const int row0 = blockIdx.x * 16;

    // Cooperative load of 16x256 X slab (float4, fully coalesced)
    const float4* src4 = (const float4*)(X + (size_t)row0 * HIDDEN);
    float4* dst4 = (float4*)Xs;
    for (int i = threadIdx.x; i < 16 * HIDDEN / 4; i += 512) dst4[i] = src4[i];
    __syncthreads();

    const int wave = threadIdx.x >> 5;           // 0..15 -> column tile
    const int lane = threadIdx.x & 31;
    const int col0 = wave * 16;
    const int m    = lane & 15;                  // A row / B col / C col
    const int half = lane >> 4;                  // lane-half selects K pair

    v8f c = {};                                  // f32 accumulator (8 VGPRs)

    for (int k0 = 0; k0 < HIDDEN; k0 += 4) {
        const int ka = k0 + 2 * half;
        // A 16x4: lane<16 holds K={k0,k0+1}, lane>=16 holds K={k0+2,k0+3}
        v2f a;
        a.x = Xs[m * HIDDEN + ka];
        a.y = Xs[m * HIDDEN + ka + 1];
        // B 4x16: N striped across lanes 0..15, lane-half selects K pair
        v2f b;
        b.x = W[(size_t)ka * HIDDEN + col0 + m];
        b.y = W[(size_t)(ka + 1) * HIDDEN + col0 + m];
        c = __builtin_amdgcn_wmma_f32_16x16x4_f32(false, a, false, b,
                                                  (short)0, c, false, false);
    }

    // D layout: VGPR j -> row (j + 8*half), col = lane&15
    const int col = col0 + m;
    float bv = 0.0f;
    if (BIAS_RELU) bv = bias[col];
#pragma unroll
    for (int j = 0; j < 8; ++j) {
        float v = c[j];
        if (BIAS_RELU) v = fmaxf(v + bv, 0.0f);
        Y[(size_t)(row0 + j + 8 * half) * HIDDEN + col] = v;
    }
}

// ---------------------------------------------------------------------------
// Edge scatter: acc[dst] += y[src] * dis[src]*dis[dst]
// 64 threads per edge, float4 loads, 4 fp32 global atomics each.
// ---------------------------------------------------------------------------
__global__ __launch_bounds__(256) void scatter_kernel(const int* __restrict__ src,
                                                      const int* __restrict__ dst,
                                                      const float* __restrict__ dis,
                                                      const float* __restrict__ y,
                                                      float* __restrict__ acc, int E) {
    int t = blockIdx.x * 256 + threadIdx.x;
    int e = t >> 6;
    if (e >= E) return;
    int c = (t & 63) * 4;
    int s = src[e], d = dst[e];
    float norm = dis[s] * dis[d];
    float4 v = *(const float4*)(y + (size_t)s * HIDDEN + c);
    float* p = acc + (size_t)d * HIDDEN + c;
    atomicAdd(p + 0, v.x * norm);
    atomicAdd(p + 1, v.y * norm);
    atomicAdd(p + 2, v.z * norm);
    atomicAdd(p + 3, v.w * norm);
}

// ---------------------------------------------------------------------------
// Layer finalize: x_out = relu(acc + y * dis^2 + b)
// ---------------------------------------------------------------------------
__global__ __launch_bounds__(256) void finalize_kernel(const float* __restrict__ acc,
                                                       const float* __restrict__ y,
                                                       const float* __restrict__ dis,
                                                       const float* __restrict__ b,
                                                       float* __restrict__ xout, int N) {
    int t = blockIdx.x * 256 + threadIdx.x;
    int n = t >> 6;
    if (n >= N) return;
    int c = (t & 63) * 4;
    float d2 = dis[n] * dis[n];
    float4 a = *(const float4*)(acc + (size_t)n * HIDDEN + c);
    float4 yy = *(const float4*)(y + (size_t)n * HIDDEN + c);
    float4 r;
    r.x = fmaxf(a.x + yy.x * d2 + b[c + 0], 0.0f);
    r.y = fmaxf(a.y + yy.y * d2 + b[c + 1], 0.0f);
    r.z = fmaxf(a.z + yy.z * d2 + b[c + 2], 0.0f);
    r.w = fmaxf(a.w + yy.w * d2 + b[c + 3], 0.0f);
    *(float4*)(xout + (size_t)n * HIDDEN + c) = r;
}

// ---------------------------------------------------------------------------
// Graph mean pooling (sum + count via atomics, then divide)
// ---------------------------------------------------------------------------
__global__ __launch_bounds__(256) void pool_kernel(const float* __restrict__ emb,
                                                   const int* __restrict__ batch,
                                                   float* __restrict__ ge_sum,
                                                   float* __restrict__ counts, int N) {
    int t = blockIdx.x * 256 + threadIdx.x;
    int n = t >> 6;
    if (n >= N) return;
    int c = (t & 63) * 4;
    int g = batch[n];
    float4 v = *(const float4*)(emb + (size_t)n * HIDDEN + c);
    float* p = ge_sum + (size_t)g * HIDDEN + c;
    atomicAdd(p + 0, v.x);
    atomicAdd(p + 1, v.y);
    atomicAdd(p + 2, v.z);
    atomicAdd(p + 3, v.w);
    if (c == 0) atomicAdd(&counts[g], 1.0f);
}

__global__ __launch_bounds__(256) void pool_div_kernel(float* __restrict__ ge,
                                                       const float* __restrict__ counts) {
    int t = blockIdx.x * 256 + threadIdx.x;
    if (t >= N_GRAPHS * HIDDEN) return;
    ge[t] = ge[t] / fmaxf(counts[t >> 8], 1.0f);
}

// ---------------------------------------------------------------------------
// Node logits: out[n] = h1[n,:] . Wn2 + bn2   (one wave32 per node)
// ---------------------------------------------------------------------------
__global__ __launch_bounds__(256) void logits_kernel(const float* __restrict__ h1,
                                                     const float* __restrict__ Wn2,
                                                     const float* __restrict__ bn2,
                                                     float* __restrict__ out, int N) {
    int w = (blockIdx.x * 256 + threadIdx.x) >> 5;
    int lane = threadIdx.x & 31;
    if (w >= N) return;
    const float* row = h1 + (size_t)w * HIDDEN;
    float s = 0.0f;
#pragma unroll
    for (int i = lane; i < HIDDEN; i += 32) s += row[i] * Wn2[i];
    for (int off = 16; off > 0; off >>= 1) s += __shfl_down(s, off, 32);
    if (lane == 0) out[w] = s + bn2[0];
}

// ---------------------------------------------------------------------------
// Value head: val[g] = relu(ge[g] @ Wv1 + bv1) @ Wv2 + bv2  (one block/graph)
// ---------------------------------------------------------------------------
__global__ __launch_bounds__(256) void value_kernel(const float* __restrict__ ge,
                                                    const float* __restrict__ Wv1,
                                                    const float* __restrict__ bv1,
                                                    const float* __restrict__ Wv2,
                                                    const float* __restrict__ bv2,
                                                    float* __restrict__ val) {
    __shared__ float gs[HIDDEN];
    __shared__ float red[256];
    int g = blockIdx.x;
    int h = threadIdx.x;
    gs[h] = ge[(size_t)g * HIDDEN + h];
    __syncthreads();
    float s = 0.0f;
    for (int k = 0; k < HIDDEN; ++k) s += gs[k] * Wv1[(size_t)k * HIDDEN + h];
    s = fmaxf(s + bv1[h], 0.0f);
    red[h] = s * Wv2[h];
    __syncthreads();
    for (int off = 128; off > 0; off >>= 1) {
        if (h < off) red[h] += red[h + off];
        __syncthreads();
    }
    if (h == 0) val[g] = red[0] + bv2[0];
}

// ---------------------------------------------------------------------------
extern "C" void kernel_launch(void* const* d_in, const int* in_sizes, int n_in,
                              void* d_out, int out_size, void* d_ws, size_t ws_size,
                              hipStream_t stream) {
    const float* coords = (const float*)d_in[0];
    const int*   eidx   = (const int*)d_in[1];     // [2, E]
    const int*   batch  = (const int*)d_in[2];
    const float* W_in   = (const float*)d_in[3];
    const float* b_in   = (const float*)d_in[4];
    const float* Wg     = (const float*)d_in[5];   // [3, H, H]
    const float* bg     = (const float*)d_in[6];   // [3, H]
    const float* Wn1    = (const float*)d_in[7];
    const float* bn1    = (const float*)d_in[8];
    const float* Wn2    = (const float*)d_in[9];
    const float* bn2    = (const float*)d_in[10];
    const float* Wv1    = (const float*)d_in[11];
    const float* bv1    = (const float*)d_in[12];
    const float* Wv2    = (const float*)d_in[13];
    const float* bv2    = (const float*)d_in[14];

    const int N = N_NODES, E = N_EDGES, H = HIDDEN, G = N_GRAPHS;
    const size_t NH = (size_t)N * H;

    const int* e_src = eidx;
    const int* e_dst = eidx + E;

    // Workspace layout
    float* ws     = (float*)d_ws;
    float* x      = ws;            // N*H
    float* y      = x + NH;        // N*H
    float* acc    = y + NH;        // N*H
    float* dis    = acc + NH;      // N   (deg -> deg_inv_sqrt in place)
    float* counts = dis + N;       // G

    // Output layout: [logits N][graph_emb G*H][node_emb N*H][value G]
    float* out    = (float*)d_out;
    float* logits = out;
    float* ge     = out + N;
    float* emb    = ge + (size_t)G * H;
    float* value  = emb + NH;

    // Degree / normalization
    hipMemsetAsync(dis, 0, N * sizeof(float), stream);
    deg_kernel<<<(E + 255) / 256, 256, 0, stream>>>(e_dst, dis, E);
    dis_kernel<<<(N + 255) / 256, 256, 0, stream>>>(dis, N);

    // Input projection
    input_proj_kernel<<<(N * H + 255) / 256, 256, 0, stream>>>(coords, W_in, b_in, x, N);

    // 3 GCN layers
    for (int l = 0; l < 3; ++l) {
        gemm_wmma_kernel<false><<<N / 16, 512, 0, stream>>>(
            x, Wg + (size_t)l * H * H, nullptr, y);
        hipMemsetAsync(acc, 0, NH * sizeof(float), stream);
        scatter_kernel<<<(E * 64) / 256, 256, 0, stream>>>(e_src, e_dst, dis, y, acc, E);
        float* xout = (l == 2) ? emb : x;
        finalize_kernel<<<(N * 64) / 256, 256, 0, stream>>>(
            acc, y, dis, bg + (size_t)l * H, xout, N);
    }

    // Graph mean pooling (into d_out graph_embedding slot)
    hipMemsetAsync(ge, 0, (size_t)G * H * sizeof(float), stream);
    hipMemsetAsync(counts, 0, G * sizeof(float), stream);
    pool_kernel<<<(N * 64) / 256, 256, 0, stream>>>(emb, batch, ge, counts, N);
    pool_div_kernel<<<(G * H + 255) / 256, 256, 0, stream>>>(ge, counts);

    // Node head: h1 = relu(emb @ Wn1 + bn1) into acc, then logits
    gemm_wmma_kernel<true><<<N / 16, 512, 0, stream>>>(emb, Wn1, bn1, acc);
    logits_kernel<<<(N * 32) / 256, 256, 0, stream>>>(acc, Wn2, bn2, logits, N);

    // Value head
    value_kernel<<<G, 256, 0, stream>>>(ge, Wv1, bv1, Wv2, bv2, value);
}